// HyperEmbedder_84628035600595
// MI455X (gfx1250) — compile-verified
//
#include <hip/hip_runtime.h>
#include <hip/hip_bf16.h>
#include <math.h>

// ---------------------------------------------------------------------------
// HyperEmbedder loss pipeline for gfx1250 (MI455X), wave32 + WMMA f32 +
// async global->LDS weight staging (ASYNCcnt) + prefetch.
//
// B=4096, P=2048, D_IN=512, ENC=[512,256,128,64,32], DEC=[32,64,128,256,512]
// TEMP=0.2, CURV c=1.0, EPS=1e-8
// ---------------------------------------------------------------------------

#define HB 4096          // batch
#define HP 2048          // parents
#define HD 512           // clip dim
#define HTEMP 0.2
#define HEPS 1e-8
#define KC 64            // K-chunk staged in LDS per iteration

typedef float v2f __attribute__((ext_vector_type(2)));
typedef float v8f __attribute__((ext_vector_type(8)));

// ---------------------------------------------------------------------------
// fp32 WMMA GEMM:  out[M,N] = act( in[M,K] @ W[K,N] + bias[N] )
//
// Block = 128 threads (4 waves). All 4 waves share one 16-col N-tile and
// cover 4 consecutive 16-row M-tiles. The weight panel W[k0:k0+KC, 16] is
// staged into LDS once per block per K-chunk via GLOBAL_LOAD_ASYNC_TO_LDS_B32
// (GVS addressing: saddr = W base, vaddr = byte offset), awaited with
// s_wait_asynccnt, then each wave runs KC/4 V_WMMA_F32_16X16X4_F32 ops with
// B fragments served from LDS (4x global-traffic reduction on weights).
//
// Assumed VGPR layouts (CDNA5 ISA 7.12.2):
//   A 16x4 : lane l holds row M=l%16; vgpr v holds K = 2*(l>=16) + v
//   B 4x16 : lane l holds col N=l%16; vgpr v holds K = 2*(l>=16) + v
//   C 16x16: lane l holds col N=l%16; vgpr r holds row M = r + 8*(l>=16)
// ---------------------------------------------------------------------------
__global__ void wmma_gemm_bias_act(const float* __restrict__ A,
                                   const float* __restrict__ W,
                                   const float* __restrict__ bias,
                                   float* __restrict__ out,
                                   int M, int K, int N, int do_gelu) {
  __shared__ float ldsW[KC * 16];            // 4 KB weight panel

  const int tid    = threadIdx.x;            // 0..127
  const int waveId = tid >> 5;               // M-tile within block
  const int lane   = tid & 31;
  const int half   = lane >> 4;              // 0: K 0..1, 1: K 2..3
  const int l16    = lane & 15;

  const int tilesN = N >> 4;
  const int blockM = blockIdx.x / tilesN;    // group of 4 M-tiles
  const int tn     = blockIdx.x - blockM * tilesN;
  const int tm     = (blockM << 2) + waveId;

  const int row = (tm << 4) + l16;           // A row owned by this lane
  const int col = (tn << 4) + l16;           // B/C column owned by this lane

  const int kc = (K < KC) ? K : KC;          // K multiple of kc for all layers
  const unsigned ldsBase = (unsigned)(size_t)(&ldsW[0]);

  v8f c = {};
  for (int k0 = 0; k0 < K; k0 += kc) {
    // ---- async-stage W[k0:k0+kc, tn*16 .. +16] into LDS ------------------
    // kc*16 in {512,1024} elements == exact multiple of 128 threads: no
    // divergence, EXEC stays all-1s (WMMA requirement).
    for (int e = tid; e < kc * 16; e += 128) {
      const int kk = e >> 4;                 // 0..kc-1
      const int nn = e & 15;
      const unsigned ldsAddr = ldsBase + (unsigned)(e * 4);
      const unsigned gOff =
          (unsigned)(((size_t)(k0 + kk) * (size_t)N + (size_t)((tn << 4) + nn)) * 4);
      asm volatile("global_load_async_to_lds_b32 %0, %1, %2 offset:0"
                   :: "v"(ldsAddr), "v"(gOff), "s"(W) : "memory");
    }
    asm volatile("s_wait_asynccnt 0x0" ::: "memory");
    __syncthreads();                         // all waves' panels landed

    // prefetch next A chunk for this lane's row (streaming operand)
    if (k0 + kc < K)
      __builtin_prefetch(A + (size_t)row * K + k0 + kc, 0, 0);

    // ---- WMMA over the staged panel --------------------------------------
    for (int kk = 0; kk < kc; kk += 4) {
      const int kb = kk + (half << 1);
      v2f a = *(const v2f*)(A + (size_t)row * K + k0 + kb);  // contiguous b64
      v2f b;
      b.x = ldsW[kb * 16 + l16];             // ds_load from staged panel
      b.y = ldsW[(kb + 1) * 16 + l16];
      c = __builtin_amdgcn_wmma_f32_16x16x4_f32(
              /*neg_a=*/false, a, /*neg_b=*/false, b,
              /*c_mod=*/(short)0, c, /*reuse_a=*/false, /*reuse_b=*/false);
    }
    __syncthreads();                         // before next chunk overwrites LDS
  }

  const float bn = bias[col];
#pragma unroll
  for (int r = 0; r < 8; ++r) {
    float v = c[r] + bn;
    if (do_gelu) v = 0.5f * v * (1.0f + erff(v * 0.70710678118654752f)); // exact GELU
    const int m = (tm << 4) + r + (half << 3);
    out[(size_t)m * N + col] = v;
  }
}

// ---------------------------------------------------------------------------
// Fused exp_map0 (fp64) -> log_map0 (fp64->fp32), per row of x[B,32].
// One wave per row (lane == column). Produces:
//   e32 : fp32 space components of hyperboloid points (for WMMA Gram)
//   t   : fp64 time component  sqrt(1 + |e|^2)
//   xn  : fp64 euclidean norm |e|
//   y0  : fp32 log-map output (decoder input)
// ---------------------------------------------------------------------------
__global__ void hyper_map(const float* __restrict__ x,
                          float* __restrict__ e32,
                          double* __restrict__ t,
                          double* __restrict__ xn,
                          float* __restrict__ y0, int nrows) {
  const int gid  = blockIdx.x * blockDim.x + threadIdx.x;
  const int row  = gid >> 5;
  const int k    = gid & 31;
  if (row >= nrows) return;

  const double xv = (double)x[(size_t)row * 32 + k];
  double sq = xv * xv;
#pragma unroll
  for (int m = 16; m >= 1; m >>= 1) sq += __shfl_xor(sq, m, 32);
  const double norm  = sqrt(sq);
  const double s     = fmax(norm, HEPS);          // sqrt(c)=1
  const double scale = sinh(s) / s;
  const double e     = scale * xv;

  double esq = e * e;
#pragma unroll
  for (int m = 16; m >= 1; m >>= 1) esq += __shfl_xor(esq, m, 32);
  const double tt  = sqrt(1.0 + esq);             // time component
  const double xnv = sqrt(esq);                   // space norm

  const double dist0 = acosh(fmax(tt, 1.0 + HEPS));
  const double yv    = dist0 * e / fmax(xnv, HEPS);

  e32[(size_t)row * 32 + k] = (float)e;
  y0 [(size_t)row * 32 + k] = (float)yv;
  if (k == 0) { t[row] = tt; xn[row] = xnv; }
}

// ---------------------------------------------------------------------------
// Gram matrix G[p, b] = <e[parent_idx[p]], e[b]>  (space components, fp32 WMMA)
// K = 32 fixed, both operands contiguous per lane -> pure v2f global loads.
// ---------------------------------------------------------------------------
__global__ void wmma_gram(const float* __restrict__ e32,
                          const long long* __restrict__ pidx,
                          float* __restrict__ G, int P, int Bn) {
  const int wavesPerBlock = blockDim.x >> 5;
  const int wave   = blockIdx.x * wavesPerBlock + (threadIdx.x >> 5);
  const int tilesB = Bn >> 4;
  const int total  = (P >> 4) * tilesB;
  if (wave >= total) return;

  const int tp   = wave / tilesB;
  const int tb   = wave - tp * tilesB;
  const int lane = threadIdx.x & 31;
  const int half = lane >> 4;
  const int l16  = lane & 15;

  const long long src = pidx[(tp << 4) + l16];     // gathered A row
  const int bcol      = (tb << 4) + l16;

  v8f c = {};
#pragma unroll
  for (int k0 = 0; k0 < 32; k0 += 4) {
    const int kb = k0 + (half << 1);
    v2f a = *(const v2f*)(e32 + (size_t)src  * 32 + kb);
    v2f b = *(const v2f*)(e32 + (size_t)bcol * 32 + kb);
    c = __builtin_amdgcn_wmma_f32_16x16x4_f32(false, a, false, b,
                                              (short)0, c, false, false);
  }
#pragma unroll
  for (int r = 0; r < 8; ++r) {
    const int p = (tp << 4) + r + (half << 3);
    G[(size_t)p * Bn + bcol] = c[r];
  }
}

// ---------------------------------------------------------------------------
// Contrastive losses, one 256-thread block per parent row p, fp64 accumulation.
// logits: dist = -acosh(max(-xyl,1+eps)); alpha = -oxy_angle; beta = pi+alpha
// masked (exp->0) iff (j==parent_idx[p] || !keep[parent_idx[p]][j]) && j!=gp
// row loss = log(sum exp(l/T)) - l_gp/T ; atomically summed into acc[1..3].
// ---------------------------------------------------------------------------
__global__ void contrastive_rows(const float* __restrict__ G,
                                 const double* __restrict__ t,
                                 const double* __restrict__ xn,
                                 const long long* __restrict__ pidx,
                                 const long long* __restrict__ gpid,
                                 const unsigned char* __restrict__ keep,
                                 double* __restrict__ acc, int P, int Bn) {
  const int p  = blockIdx.x;
  const int tid = threadIdx.x;
  const long long pr = pidx[p];
  const long long gp = gpid[p];
  const double tp    = t[pr];
  const double invT  = 1.0 / HTEMP;
  const double PI_D  = 3.14159265358979323846;

  __shared__ double gv[3];
  __shared__ double rd[256], ra[256], rb[256];

  double sd = 0.0, sa = 0.0, sb = 0.0;
  for (int j = tid; j < Bn; j += blockDim.x) {
    const double tj  = t[j];
    const double xyl = (double)G[(size_t)p * Bn + j] - tp * tj;   // Lorentz ip
    // distance logit
    const double dlog = -acosh(fmax(-xyl, 1.0 + HEPS));
    // oxy angle logit (x=embedded[j], y=parent[p])
    const double numer  = tp + xyl * tj;
    const double denom  = xn[j] * sqrt(fmax(xyl * xyl - 1.0, 0.0));
    double cosang = numer / (denom + HEPS);
    cosang = fmin(fmax(cosang, -1.0 + HEPS), 1.0 - HEPS);
    const double alog = -acos(cosang);
    const double blog = PI_D + alog;

    const bool masked =
        ((j == (int)pr) || (keep[(size_t)pr * Bn + j] == 0)) && (j != (int)gp);
    if (!masked) {
      sd += exp(dlog * invT);
      sa += exp(alog * invT);
      sb += exp(blog * invT);
    }
    if (j == (int)gp) { gv[0] = dlog; gv[1] = alog; gv[2] = blog; }
  }

  rd[tid] = sd; ra[tid] = sa; rb[tid] = sb;
  __syncthreads();
  for (int s = 128; s > 0; s >>= 1) {
    if (tid < s) { rd[tid] += rd[tid + s]; ra[tid] += ra[tid + s]; rb[tid] += rb[tid + s]; }
    __syncthreads();
  }
  if (tid == 0) {
    atomicAdd(&acc[1], log(rd[0]) - gv[0] * invT);
    atomicAdd(&acc[2], log(ra[0]) - gv[1] * invT);
    atomicAdd(&acc[3], log(rb[0]) - gv[2] * invT);
  }
}

// ---------------------------------------------------------------------------
// Recon loss: sum (rec - clip)^2 in fp64 -> acc[0]
// ---------------------------------------------------------------------------
__global__ void recon_sq(const float* __restrict__ rec,
                         const float* __restrict__ clip,
                         double* __restrict__ acc, int n) {
  double local = 0.0;
  for (int i = blockIdx.x * blockDim.x + threadIdx.x; i < n;
       i += gridDim.x * blockDim.x) {
    const double d = (double)rec[i] - (double)clip[i];
    local += d * d;
  }
  __shared__ double sm[256];
  sm[threadIdx.x] = local;
  __syncthreads();
  for (int s = 128; s > 0; s >>= 1) {
    if (threadIdx.x < s) sm[threadIdx.x] += sm[threadIdx.x + s];
    __syncthreads();
  }
  if (threadIdx.x == 0) atomicAdd(&acc[0], sm[0]);
}

__global__ void init_acc(double* acc) {
  if (threadIdx.x < 4) acc[threadIdx.x] = 0.0;
}

__global__ void finalize(const double* __restrict__ acc, float* __restrict__ out,
                         int recon_n, int P) {
  const double recon = acc[0] / (double)recon_n;
  const double ld = acc[1] / (double)P;
  const double la = acc[2] / (double)P;
  const double lb = acc[3] / (double)P;
  const double contrast = ld + 0.5 * (lb + la);
  out[0] = (float)(recon + contrast);
  out[1] = (float)recon;
  out[2] = (float)contrast;
}

// ---------------------------------------------------------------------------
// Host-side orchestration
// ---------------------------------------------------------------------------
static inline void launch_gemm(const float* A, const float* W, const float* b,
                               float* out, int M, int K, int N, int gelu,
                               hipStream_t stream) {
  const int blocks = (M >> 6) * (N >> 4);    // 4 M-tiles per block
  wmma_gemm_bias_act<<<blocks, 128, 0, stream>>>(A, W, b, out, M, K, N, gelu);
}

extern "C" void kernel_launch(void* const* d_in, const int* in_sizes, int n_in,
                              void* d_out, int out_size, void* d_ws, size_t ws_size,
                              hipStream_t stream) {
  (void)in_sizes; (void)n_in; (void)out_size; (void)ws_size;

  // ---- inputs (setup_inputs dict order, pytrees flattened leaf-wise) ----
  const float* clip = (const float*)d_in[0];
  const float* encW[5]; const float* encB[5];
  const float* decW[5]; const float* decB[5];
  for (int i = 0; i < 5; ++i) {
    encW[i] = (const float*)d_in[1 + 2 * i];
    encB[i] = (const float*)d_in[2 + 2 * i];
    decW[i] = (const float*)d_in[11 + 2 * i];
    decB[i] = (const float*)d_in[12 + 2 * i];
  }
  const long long*     pidx = (const long long*)d_in[21];  // int64
  const long long*     gpid = (const long long*)d_in[22];  // int64
  const unsigned char* keep = (const unsigned char*)d_in[23]; // bool (B,B)
  float* out = (float*)d_out;

  // ---- workspace carve-out (~51.5 MB) ----
  char* ws = (char*)d_ws;
  size_t off = 0;
  auto carve = [&](size_t bytes) -> char* {
    char* p = ws + off;
    off += (bytes + 255) & ~(size_t)255;
    return p;
  };
  float*  actA = (float*)carve((size_t)HB * HD * 4);   // 8 MB ping
  float*  actB = (float*)carve((size_t)HB * HD * 4);   // 8 MB pong
  float*  e32  = (float*)carve((size_t)HB * 32 * 4);   // hyperboloid space (fp32)
  float*  y0   = (float*)carve((size_t)HB * 32 * 4);   // log-map output
  double* tim  = (double*)carve((size_t)HB * 8);       // time components
  double* xnrm = (double*)carve((size_t)HB * 8);       // space norms
  double* acc  = (double*)carve(4 * 8);                // loss accumulators
  float*  G    = (float*)carve((size_t)HP * HB * 4);   // 32 MB Gram matrix

  init_acc<<<1, 32, 0, stream>>>(acc);

  // ---- encoder: Linear, (GELU, Linear)x4 -> apply GELU on outputs 0..3 ----
  launch_gemm(clip, encW[0], encB[0], actA, HB, 512, 512, 1, stream);
  launch_gemm(actA, encW[1], encB[1], actB, HB, 512, 256, 1, stream);
  launch_gemm(actB, encW[2], encB[2], actA, HB, 256, 128, 1, stream);
  launch_gemm(actA, encW[3], encB[3], actB, HB, 128,  64, 1, stream);
  launch_gemm(actB, encW[4], encB[4], actA, HB,  64,  32, 0, stream); // raw 32-d

  // ---- fp64 exp_map0 + log_map0 (fused) ----
  hyper_map<<<(HB * 32) / 256, 256, 0, stream>>>(actA, e32, tim, xnrm, y0, HB);

  // ---- decoder: (Linear, GELU)x4, Linear ----
  launch_gemm(y0,   decW[0], decB[0], actB, HB,  32,  64, 1, stream);
  launch_gemm(actB, decW[1], decB[1], actA, HB,  64, 128, 1, stream);
  launch_gemm(actA, decW[2], decB[2], actB, HB, 128, 256, 1, stream);
  launch_gemm(actB, decW[3], decB[3], actA, HB, 256, 512, 1, stream);
  launch_gemm(actA, decW[4], decB[4], actB, HB, 512, 512, 0, stream); // reconstructed

  // ---- recon MSE ----
  recon_sq<<<512, 256, 0, stream>>>(actB, clip, acc, HB * HD);

  // ---- pairwise Gram via WMMA, then fp64 contrastive rows ----
  {
    const int tiles  = (HP >> 4) * (HB >> 4);
    wmma_gram<<<tiles / 4, 128, 0, stream>>>(e32, pidx, G, HP, HB);
  }
  contrastive_rows<<<HP, 256, 0, stream>>>(G, tim, xnrm, pidx, gpid, keep,
                                           acc, HP, HB);

  // ---- combine ----
  finalize<<<1, 1, 0, stream>>>(acc, out, HB * HD, HP);
}